// Mamba2_70995809402924
// MI455X (gfx1250) — compile-verified
//
#include <hip/hip_runtime.h>
#include <hip/hip_bf16.h>
#include <math.h>

#define D_MODEL   1024
#define D_STATE   128
#define D_CONV    4
#define HEADDIM   64
#define D_INNER   2048
#define NHEADS    32
#define D_IN_PROJ 4384      // 2*D_INNER + 2*D_STATE + NHEADS
#define CONV_DIM  2304      // D_INNER + 2*D_STATE
#define BATCH     2
#define SEQLEN    512
#define EPS_RMS   1e-5f

typedef __attribute__((ext_vector_type(2))) float v2f;
typedef __attribute__((ext_vector_type(8))) float v8f;

// Async gather of one dword from global into LDS (CDNA5 GLOBAL_LOAD_ASYNC_TO_LDS_B32,
// tracked by ASYNCcnt). lds = 0-based LDS byte offset (low 32 bits of flat LDS addr).
__device__ __forceinline__ void async_ld_b32(unsigned lds, const float* gp) {
    asm volatile("global_load_async_to_lds_b32 %0, %1, off"
                 :: "v"(lds), "v"(gp) : "memory");
}
__device__ __forceinline__ void wait_async0() {
    asm volatile("s_wait_asynccnt 0x0" ::: "memory");
}

// ---------------------------------------------------------------------------
// GEMM:  C[m,n] = sum_k A[m,k] * W[n,k]   (A: MxK row-major, W: NxK row-major)
// Block tile 64x128, BK=16, double-buffered LDS fed by async-to-LDS.
// 256 threads = 8 waves (4x2); each wave owns a 16x64 slab = four 16x16 f32
// WMMA accumulators sharing one A fragment.
// Both LDS tiles keep K innermost with +2 pad: fragment (k,k+1) pairs are
// contiguous and 8B-aligned -> single ds_load_b64 per fragment, no v_movs,
// and stride-18 rows are bank-conflict-free.
// ---------------------------------------------------------------------------
#define BM 64
#define BN 128
#define BK 16
#define KP (BK + 2)

__global__ __launch_bounds__(256) void gemm_nt_wmma(
    const float* __restrict__ A, const float* __restrict__ W,
    float* __restrict__ C, int M, int N, int K)
{
    __shared__ float As[2][BM][KP];    // As[buf][m][k]
    __shared__ float Bs[2][BN][KP];    // Bs[buf][n][k] = W[n][k]

    const int tid   = threadIdx.x;
    const int wave  = tid >> 5;
    const int lane  = tid & 31;
    const int wm    = wave >> 1;       // 0..3  (16-row slab)
    const int wn    = wave & 1;        // 0..1  (64-col slab)
    const int lhalf = lane >> 4;       // 0/1 -> K pair within 16x16x4 frag
    const int lmod  = lane & 15;

    const int block_m = blockIdx.y * BM;
    const int block_n = blockIdx.x * BN;
    const bool nfull  = (block_n + BN) <= N;

    v8f acc[4] = {{}, {}, {}, {}};

    const int sr = tid >> 4;           // 0..15 staging row base
    const int sc = tid & 15;           // 0..15 staging k index

    auto stage_tile = [&](int k0, int buf) {
        // A tile 64x16 (always in-range: M multiple of 64)
        #pragma unroll
        for (int i = 0; i < 4; ++i) {
            int r = sr + i * 16;
            async_ld_b32((unsigned)(size_t)&As[buf][r][sc],
                         &A[(size_t)(block_m + r) * K + k0 + sc]);
        }
        // W tile 128x16 into Bs[n][k]
        if (nfull) {
            #pragma unroll
            for (int i = 0; i < 8; ++i) {
                int n = sr + i * 16;
                async_ld_b32((unsigned)(size_t)&Bs[buf][n][sc],
                             &W[(size_t)(block_n + n) * K + k0 + sc]);
            }
        } else {
            #pragma unroll
            for (int i = 0; i < 8; ++i) {
                int n  = sr + i * 16;
                int gn = block_n + n;
                Bs[buf][n][sc] = (gn < N) ? W[(size_t)gn * K + k0 + sc] : 0.0f;
            }
        }
    };

    stage_tile(0, 0);
    wait_async0();
    __syncthreads();

    int cur = 0;
    const int m = wm * 16 + lmod;
    for (int k0 = 0; k0 < K; k0 += BK) {
        if (k0 + BK < K) stage_tile(k0 + BK, cur ^ 1);   // overlap with compute

        #pragma unroll
        for (int kk = 0; kk < BK; kk += 4) {
            // a[r] = A[m, 2*lhalf + r]; contiguous + 8B aligned -> ds_load_b64
            v2f a = *(const v2f*)&As[cur][m][kk + 2 * lhalf];
            #pragma unroll
            for (int s = 0; s < 4; ++s) {
                int n = wn * 64 + s * 16 + lmod;
                v2f b = *(const v2f*)&Bs[cur][n][kk + 2 * lhalf];
                acc[s] = __builtin_amdgcn_wmma_f32_16x16x4_f32(
                    false, a, false, b, (short)0, acc[s], false, false);
            }
        }
        wait_async0();
        __syncthreads();
        cur ^= 1;
    }

    // D layout: VGPR r holds M=r (lanes 0-15) / M=r+8 (lanes 16-31), N=lane%16
    #pragma unroll
    for (int s = 0; s < 4; ++s) {
        #pragma unroll
        for (int r = 0; r < 8; ++r) {
            int mm = block_m + wm * 16 + r + 8 * lhalf;
            int n  = block_n + wn * 64 + s * 16 + lmod;
            if (n < N) C[(size_t)mm * N + n] = acc[s][r];
        }
    }
}

// ---------------------------------------------------------------------------
// Depthwise causal conv1d(k=4) + bias + SiLU over xBC channels, plus
// softplus(dt + dt_bias). One block per token.
// ---------------------------------------------------------------------------
__global__ __launch_bounds__(256) void conv_silu_dt(
    const float* __restrict__ zxbcdt, const float* __restrict__ conv_w,
    const float* __restrict__ conv_b, const float* __restrict__ dt_bias,
    float* __restrict__ xBC_act, float* __restrict__ dt_proc)
{
    const int token = blockIdx.x;          // b*SEQLEN + l
    const int l     = token % SEQLEN;
    const int tid   = threadIdx.x;

    for (int c = tid; c < CONV_DIM; c += 256) {
        float acc = conv_b[c];
        #pragma unroll
        for (int j = 0; j < D_CONV; ++j) {
            int lt = l - (D_CONV - 1) + j;
            if (lt >= 0) {
                acc += conv_w[c * D_CONV + j] *
                       zxbcdt[(size_t)(token - (l - lt)) * D_IN_PROJ + D_INNER + c];
            }
        }
        // SiLU
        xBC_act[(size_t)token * CONV_DIM + c] = acc / (1.0f + __expf(-acc));
    }

    if (tid < NHEADS) {
        float v = zxbcdt[(size_t)token * D_IN_PROJ + D_INNER + CONV_DIM + tid]
                + dt_bias[tid];
        float sp = (v > 20.0f) ? v : log1pf(__expf(v));
        dt_proc[token * NHEADS + tid] = sp;
    }
}

// ---------------------------------------------------------------------------
// Selective scan. One workgroup per (batch, head). State (64 x 128) lives in
// registers: thread t owns p = t%64, n in [32*(t/64), +32). B_t/C_t for step
// t+1 stream into the alternate LDS buffer via async-to-LDS while step t's
// FMAs run; x/dt for t+1 prefetch into registers. One wait+barrier per step.
// ---------------------------------------------------------------------------
__global__ __launch_bounds__(256) void scan_kernel(
    const float* __restrict__ xBC_act, const float* __restrict__ dt_proc,
    const float* __restrict__ A_log, const float* __restrict__ D_param,
    float* __restrict__ y)
{
    const int b   = blockIdx.x / NHEADS;
    const int h   = blockIdx.x % NHEADS;
    const int tid = threadIdx.x;
    const int p   = tid & 63;
    const int n0  = (tid >> 6) * 32;

    const float Aneg = -__expf(A_log[h]);
    const float Dp   = D_param[h];

    float s[32];
    #pragma unroll
    for (int j = 0; j < 32; ++j) s[j] = 0.0f;

    __shared__ float Bs[2][D_STATE];
    __shared__ float Cs[2][D_STATE];
    __shared__ float yred[256];

    // Per-thread staging target per buffer: tid<128 -> Bs, else Cs.
    // (B and C are contiguous in xBC_act: offset D_INNER + tid covers both.)
    const unsigned dst0 = (tid < D_STATE)
                        ? (unsigned)(size_t)&Bs[0][tid]
                        : (unsigned)(size_t)&Cs[0][tid - D_STATE];
    const unsigned dst1 = (tid < D_STATE)
                        ? (unsigned)(size_t)&Bs[1][tid]
                        : (unsigned)(size_t)&Cs[1][tid - D_STATE];

    // Prime step 0
    size_t base = (size_t)(b * SEQLEN) * CONV_DIM;
    async_ld_b32(dst0, &xBC_act[base + D_INNER + tid]);
    float x_cur  = xBC_act[base + h * HEADDIM + p];
    float dt_cur = dt_proc[(b * SEQLEN) * NHEADS + h];
    wait_async0();
    __syncthreads();

    int cur = 0;
    for (int t = 0; t < SEQLEN; ++t) {
        float x_nxt = 0.0f, dt_nxt = 0.0f;
        if (t + 1 < SEQLEN) {
            const size_t basen = (size_t)(b * SEQLEN + t + 1) * CONV_DIM;
            async_ld_b32(cur ? dst0 : dst1, &xBC_act[basen + D_INNER + tid]);
            x_nxt  = xBC_act[basen + h * HEADDIM + p];
            dt_nxt = dt_proc[(b * SEQLEN + t + 1) * NHEADS + h];
        }

        const float dA  = __expf(dt_cur * Aneg);
        const float dtx = dt_cur * x_cur;

        float py = 0.0f;
        #pragma unroll
        for (int j = 0; j < 32; ++j) {
            s[j] = s[j] * dA + dtx * Bs[cur][n0 + j];
            py  += s[j] * Cs[cur][n0 + j];
        }
        yred[tid] = py;
        __syncthreads();

        if (tid < HEADDIM) {   // p == tid here, x_cur matches this p
            float yv = yred[tid] + yred[tid + 64] + yred[tid + 128] + yred[tid + 192]
                     + Dp * x_cur;
            y[(size_t)(b * SEQLEN + t) * D_INNER + h * HEADDIM + tid] = yv;
        }

        wait_async0();         // next step's B/C landed
        __syncthreads();       // protect yred + publish next buffer
        x_cur  = x_nxt;
        dt_cur = dt_nxt;
        cur ^= 1;
    }
}

// ---------------------------------------------------------------------------
// Gated RMSNorm: yg = y*silu(z); yn = yg * rsqrt(mean(yg^2)+eps) * norm_w.
// One block per token; 8 values per thread held in registers.
// ---------------------------------------------------------------------------
__global__ __launch_bounds__(256) void gated_rmsnorm(
    const float* __restrict__ y, const float* __restrict__ zxbcdt,
    const float* __restrict__ norm_w, float* __restrict__ yn)
{
    const int token = blockIdx.x;
    const int tid   = threadIdx.x;

    float g[8];
    float ss = 0.0f;
    #pragma unroll
    for (int j = 0; j < 8; ++j) {
        int i    = tid + j * 256;
        float zv = zxbcdt[(size_t)token * D_IN_PROJ + i];   // z = first D_INNER cols
        float yv = y[(size_t)token * D_INNER + i];
        float gv = yv * (zv / (1.0f + __expf(-zv)));
        g[j] = gv;
        ss  += gv * gv;
    }

    __shared__ float red[256];
    red[tid] = ss;
    __syncthreads();
    for (int s2 = 128; s2 > 0; s2 >>= 1) {
        if (tid < s2) red[tid] += red[tid + s2];
        __syncthreads();
    }
    const float scale = rsqrtf(red[0] / (float)D_INNER + EPS_RMS);

    #pragma unroll
    for (int j = 0; j < 8; ++j) {
        int i = tid + j * 256;
        yn[(size_t)token * D_INNER + i] = g[j] * scale * norm_w[i];
    }
}

// ---------------------------------------------------------------------------
// Launch
// ---------------------------------------------------------------------------
extern "C" void kernel_launch(void* const* d_in, const int* in_sizes, int n_in,
                              void* d_out, int out_size, void* d_ws, size_t ws_size,
                              hipStream_t stream) {
    const float* u       = (const float*)d_in[0];
    const float* W_in    = (const float*)d_in[1];
    const float* conv_w  = (const float*)d_in[2];
    const float* conv_b  = (const float*)d_in[3];
    const float* dt_bias = (const float*)d_in[4];
    const float* A_log   = (const float*)d_in[5];
    const float* D_param = (const float*)d_in[6];
    const float* norm_w  = (const float*)d_in[7];
    const float* W_out   = (const float*)d_in[8];
    float*       out     = (float*)d_out;

    const int M = BATCH * SEQLEN;                 // 1024 tokens

    // Workspace carve-out (floats): ~44 MB total
    float* ws      = (float*)d_ws;
    float* zxbcdt  = ws;                                          // M * 4384
    float* xBC_act = zxbcdt  + (size_t)M * D_IN_PROJ;             // M * 2304
    float* dt_proc = xBC_act + (size_t)M * CONV_DIM;              // M * 32
    float* ybuf    = dt_proc + (size_t)M * NHEADS;                // M * 2048
    float* ynbuf   = ybuf    + (size_t)M * D_INNER;               // M * 2048

    // 1) In-projection: zxbcdt = u @ W_in^T    (M=1024, N=4384, K=1024)
    {
        dim3 grid((D_IN_PROJ + BN - 1) / BN, M / BM);
        gemm_nt_wmma<<<grid, 256, 0, stream>>>(u, W_in, zxbcdt, M, D_IN_PROJ, D_MODEL);
    }

    // 2) Depthwise conv + SiLU + softplus(dt)
    conv_silu_dt<<<M, 256, 0, stream>>>(zxbcdt, conv_w, conv_b, dt_bias,
                                        xBC_act, dt_proc);

    // 3) Selective scan (register-resident state, double-buffered B/C)
    scan_kernel<<<BATCH * NHEADS, 256, 0, stream>>>(xBC_act, dt_proc, A_log,
                                                    D_param, ybuf);

    // 4) Gated RMSNorm
    gated_rmsnorm<<<M, 256, 0, stream>>>(ybuf, zxbcdt, norm_w, ynbuf);

    // 5) Out-projection: out = yn @ W_out^T    (M=1024, N=1024, K=2048)
    {
        dim3 grid((D_MODEL + BN - 1) / BN, M / BM);
        gemm_nt_wmma<<<grid, 256, 0, stream>>>(ynbuf, W_out, out, M, D_MODEL, D_INNER);
    }
}